// PolymerGATNN_17935783428687
// MI455X (gfx1250) — compile-verified
//
#include <hip/hip_runtime.h>
#include <hip/hip_bf16.h>

#define N_NODES   50000
#define N_EDGES   800000
#define N_GRAPHS  500
#define IN_DIM    128
#define HID       32
#define HEADS     4
#define OUT_DIM   64
#define E_TOT     (N_EDGES + N_NODES)   // edges + self loops

typedef __attribute__((ext_vector_type(2))) float v2f;
typedef __attribute__((ext_vector_type(8))) float v8f;

// ---------- order-preserving float<->uint encoding for atomicMax ----------
__device__ __forceinline__ unsigned enc_f32(float f) {
    unsigned u = __float_as_uint(f);
    return (u & 0x80000000u) ? ~u : (u | 0x80000000u);
}
__device__ __forceinline__ float dec_f32(unsigned u) {
    return (u & 0x80000000u) ? __uint_as_float(u & 0x7fffffffu)
                             : __uint_as_float(~u);
}
#define ENC_NEG_INF 0x007FFFFFu   // enc_f32(-inf)

// -------------------------- fill kernels ---------------------------------
__global__ void k_fill_f32(float* p, long long n, float v) {
    long long i = (long long)blockIdx.x * blockDim.x + threadIdx.x;
    if (i < n) p[i] = v;
}
__global__ void k_fill_u32(unsigned* p, long long n, unsigned v) {
    long long i = (long long)blockIdx.x * blockDim.x + threadIdx.x;
    if (i < n) p[i] = v;
}

// ------------------- self-loop mean edge attributes ----------------------
__global__ void k_loop_attr_accum(const int* __restrict__ dst,
                                  const float* __restrict__ ea,
                                  float* __restrict__ la, float* __restrict__ cnt) {
    int e = blockIdx.x * blockDim.x + threadIdx.x;
    if (e >= N_EDGES) return;
    int d = dst[e];
    atomicAdd(&cnt[d], 1.0f);
    atomicAdd(&la[2 * d + 0], ea[2 * e + 0]);
    atomicAdd(&la[2 * d + 1], ea[2 * e + 1]);
}
__global__ void k_loop_attr_div(float* __restrict__ la, const float* __restrict__ cnt) {
    int n = blockIdx.x * blockDim.x + threadIdx.x;
    if (n >= N_NODES) return;
    float c = fmaxf(cnt[n], 1.0f);
    la[2 * n + 0] /= c;
    la[2 * n + 1] /= c;
}

// ---- precompute em[w*H+h] = sum_d We[w, h*D+d] * a_edge[h, d] (both layers)
__global__ void k_edge_mix(const float* __restrict__ We1, const float* __restrict__ ae1,
                           const float* __restrict__ We2, const float* __restrict__ ae2,
                           float* __restrict__ em) {
    int t = threadIdx.x;
    if (t < 8) {               // layer 1: w in {0,1}, h in {0..3}
        int w = t >> 2, h = t & 3;
        float s = 0.f;
        for (int d = 0; d < HID; ++d)
            s += We1[w * (HEADS * HID) + h * HID + d] * ae1[h * HID + d];
        em[w * HEADS + h] = s;
    } else if (t < 10) {       // layer 2: w in {0,1}, h = 0
        int w = t - 8;
        float s = 0.f;
        for (int d = 0; d < OUT_DIM; ++d)
            s += We2[w * OUT_DIM + d] * ae2[d];
        em[8 + w] = s;
    }
}

// -------------------- fp32 WMMA GEMM: C[M,N] = A[M,128] @ B[128,N] -------
template <int N>
__global__ __launch_bounds__(256) void k_wmma_gemm(const float* __restrict__ A,
                                                   const float* __restrict__ B,
                                                   float* __restrict__ C, int M) {
    const int K = 128;
    const int ntiles = N / 16;
    const int mtiles = M / 16;
    int wave = blockIdx.x * (blockDim.x >> 5) + (threadIdx.x >> 5);
    if (wave >= mtiles * ntiles) return;           // wave-uniform exit
    int mt = wave / ntiles, nt = wave - mt * ntiles;
    int lane = threadIdx.x & 31;
    int hl   = lane >> 4;                          // half-wave: 0 or 1
    int l16  = lane & 15;

    const float* Arow = A + (long long)(mt * 16 + l16) * K;  // A row for this lane
    const float* Bcol = B + nt * 16 + l16;                   // B column base

    v8f c = {};
#pragma unroll
    for (int k0 = 0; k0 < K; k0 += 4) {
        // A 16x4 fragment: VGPR0 = K 0/2, VGPR1 = K 1/3 (per half-wave)
        v2f a, b;
        a.x = Arow[k0 + 2 * hl];
        a.y = Arow[k0 + 2 * hl + 1];
        // B 4x16 fragment: VGPR0 = rows k0+0 / k0+1, VGPR1 = rows k0+2 / k0+3
        b.x = Bcol[(long long)(k0 + hl) * N];
        b.y = Bcol[(long long)(k0 + 2 + hl) * N];
        c = __builtin_amdgcn_wmma_f32_16x16x4_f32(false, a, false, b,
                                                  (short)0, c, false, false);
    }
    // C/D layout: VGPR r -> row (r + 8*half), col = lane%16
    float* Crow = C + (long long)(mt * 16) * N + nt * 16 + l16;
#pragma unroll
    for (int r = 0; r < 8; ++r)
        Crow[(long long)(r + 8 * hl) * N] = c[r];
}

// ------------- per-(node,head) attention logit halves --------------------
template <int H, int D>
__global__ void k_alphas(const float* __restrict__ xp,
                         const float* __restrict__ a_src, const float* __restrict__ a_dst,
                         float* __restrict__ as, float* __restrict__ ad) {
    int idx = blockIdx.x * blockDim.x + threadIdx.x;
    if (idx >= N_NODES * H) return;
    int n = idx / H, h = idx - n * H;
    const float* xr = xp + (long long)n * (H * D) + h * D;
    float s1 = 0.f, s2 = 0.f;
    for (int d = 0; d < D; ++d) {
        float v = xr[d];
        s1 += v * a_src[h * D + d];
        s2 += v * a_dst[h * D + d];
    }
    as[idx] = s1;
    ad[idx] = s2;
}

// ---------------- per-edge alpha (with self loops, leaky relu) -----------
template <int H>
__device__ __forceinline__ float edge_alpha(int e, int h,
        const int* __restrict__ src, const int* __restrict__ dst,
        const float* __restrict__ ea, const float* __restrict__ la,
        const float* __restrict__ em, const float* __restrict__ as,
        const float* __restrict__ ad, int& s, int& d) {
    float e0, e1;
    if (e < N_EDGES) {
        s = src[e]; d = dst[e];
        e0 = ea[2 * e]; e1 = ea[2 * e + 1];
    } else {
        s = e - N_EDGES; d = s;
        e0 = la[2 * s]; e1 = la[2 * s + 1];
    }
    float a = as[s * H + h] + ad[d * H + h] + e0 * em[h] + e1 * em[H + h];
    return a > 0.f ? a : 0.2f * a;   // leaky_relu(0.2)
}

template <int H>
__global__ void k_edge_max(const int* __restrict__ src, const int* __restrict__ dst,
                           const float* __restrict__ ea, const float* __restrict__ la,
                           const float* __restrict__ em, const float* __restrict__ as,
                           const float* __restrict__ ad, unsigned* __restrict__ m) {
    long long idx = (long long)blockIdx.x * blockDim.x + threadIdx.x;
    if (idx >= (long long)E_TOT * H) return;
    int e = (int)(idx / H), h = (int)(idx - (long long)e * H);
    int s, d;
    float a = edge_alpha<H>(e, h, src, dst, ea, la, em, as, ad, s, d);
    atomicMax(&m[d * H + h], enc_f32(a));
}

template <int H>
__global__ void k_edge_sum(const int* __restrict__ src, const int* __restrict__ dst,
                           const float* __restrict__ ea, const float* __restrict__ la,
                           const float* __restrict__ em, const float* __restrict__ as,
                           const float* __restrict__ ad, const unsigned* __restrict__ m,
                           float* __restrict__ den) {
    long long idx = (long long)blockIdx.x * blockDim.x + threadIdx.x;
    if (idx >= (long long)E_TOT * H) return;
    int e = (int)(idx / H), h = (int)(idx - (long long)e * H);
    int s, d;
    float a = edge_alpha<H>(e, h, src, dst, ea, la, em, as, ad, s, d);
    atomicAdd(&den[d * H + h], __expf(a - dec_f32(m[d * H + h])));
}

template <int H, int D>
__global__ void k_edge_aggr(const int* __restrict__ src, const int* __restrict__ dst,
                            const float* __restrict__ ea, const float* __restrict__ la,
                            const float* __restrict__ em, const float* __restrict__ as,
                            const float* __restrict__ ad, const unsigned* __restrict__ m,
                            const float* __restrict__ den, const float* __restrict__ xp,
                            float* __restrict__ out) {
    const int HD = H * D;
    long long tid = (long long)blockIdx.x * blockDim.x + threadIdx.x;
    if (tid >= (long long)E_TOT * HD) return;
    int e   = (int)(tid / HD);
    int dim = (int)(tid - (long long)e * HD);
    int h   = dim / D;
    int s, d;
    float a = edge_alpha<H>(e, h, src, dst, ea, la, em, as, ad, s, d);
    float w = __expf(a - dec_f32(m[d * H + h])) / den[d * H + h];
    atomicAdd(&out[(long long)d * HD + dim], xp[(long long)s * HD + dim] * w);
}

// ------------------------- epilogues -------------------------------------
__global__ void k_bias_relu(float* __restrict__ x, const float* __restrict__ b,
                            long long n, int dim) {
    long long i = (long long)blockIdx.x * blockDim.x + threadIdx.x;
    if (i >= n) return;
    float v = x[i] + b[i % dim];
    x[i] = v > 0.f ? v : 0.f;
}

__global__ void k_pool_accum(const float* __restrict__ out2, const int* __restrict__ pm,
                             float* __restrict__ psum, float* __restrict__ pcnt) {
    int idx = blockIdx.x * blockDim.x + threadIdx.x;
    if (idx >= N_NODES * OUT_DIM) return;
    int n = idx >> 6, c = idx & 63;
    int g = pm[n];
    atomicAdd(&psum[g * OUT_DIM + c], out2[idx]);
    if (c == 0) atomicAdd(&pcnt[g], 1.0f);
}

__global__ void k_pool_final(const float* __restrict__ psum, const float* __restrict__ pcnt,
                             const float* __restrict__ b2, float* __restrict__ out) {
    int i = blockIdx.x * blockDim.x + threadIdx.x;
    if (i >= N_GRAPHS * OUT_DIM) return;
    int g = i >> 6, c = i & 63;
    out[i] = psum[i] / fmaxf(pcnt[g], 1.0f) + b2[c];
}

// -------------------------------------------------------------------------
static inline int gridFor(long long n, int b = 256) { return (int)((n + b - 1) / b); }

extern "C" void kernel_launch(void* const* d_in, const int* in_sizes, int n_in,
                              void* d_out, int out_size, void* d_ws, size_t ws_size,
                              hipStream_t stream) {
    const float* x    = (const float*)d_in[0];
    const float* ea   = (const float*)d_in[1];
    const int*   eidx = (const int*)  d_in[2];
    const int*   pm   = (const int*)  d_in[3];
    const float* W1   = (const float*)d_in[4];
    const float* We1  = (const float*)d_in[5];
    const float* as1w = (const float*)d_in[6];
    const float* ad1w = (const float*)d_in[7];
    const float* ae1  = (const float*)d_in[8];
    const float* b1   = (const float*)d_in[9];
    const float* W2   = (const float*)d_in[10];
    const float* We2  = (const float*)d_in[11];
    const float* as2w = (const float*)d_in[12];
    const float* ad2w = (const float*)d_in[13];
    const float* ae2  = (const float*)d_in[14];
    const float* b2   = (const float*)d_in[15];

    const int* src = eidx;
    const int* dst = eidx + N_EDGES;

    // ---- workspace layout (bump allocator, all f32 unless noted) ----
    float* ws = (float*)d_ws;
    size_t o = 0;
    auto alloc = [&](size_t n) { float* p = ws + o; o += n; return p; };
    float* xp1   = alloc((size_t)N_NODES * 128);
    float* h1    = alloc((size_t)N_NODES * 128);   // layer-1 output / layer-2 input
    float* xp2   = alloc((size_t)N_NODES * 64);
    float* out2  = alloc((size_t)N_NODES * 64);
    float* la    = alloc((size_t)N_NODES * 2);     // self-loop attrs (sum -> mean)
    float* cnt   = alloc((size_t)N_NODES);
    float* as1   = alloc((size_t)N_NODES * HEADS);
    float* ad1   = alloc((size_t)N_NODES * HEADS);
    float* den1  = alloc((size_t)N_NODES * HEADS);
    float* as2   = alloc((size_t)N_NODES);
    float* ad2   = alloc((size_t)N_NODES);
    float* den2  = alloc((size_t)N_NODES);
    float* em    = alloc(16);                      // [0..7]=layer1, [8..9]=layer2
    float* psum  = alloc((size_t)N_GRAPHS * OUT_DIM);
    float* pcnt  = alloc((size_t)N_GRAPHS);
    unsigned* m1 = (unsigned*)alloc((size_t)N_NODES * HEADS);
    unsigned* m2 = (unsigned*)alloc((size_t)N_NODES);

    // ---- re-initialize all accumulators (deterministic per call) ----
    k_fill_f32<<<gridFor((long long)N_NODES * 128), 256, 0, stream>>>(h1,  (long long)N_NODES * 128, 0.f);
    k_fill_f32<<<gridFor((long long)N_NODES * 64),  256, 0, stream>>>(out2,(long long)N_NODES * 64,  0.f);
    k_fill_f32<<<gridFor((long long)N_NODES * 2),   256, 0, stream>>>(la,  (long long)N_NODES * 2,   0.f);
    k_fill_f32<<<gridFor(N_NODES), 256, 0, stream>>>(cnt,  N_NODES, 0.f);
    k_fill_f32<<<gridFor((long long)N_NODES * HEADS), 256, 0, stream>>>(den1, (long long)N_NODES * HEADS, 0.f);
    k_fill_f32<<<gridFor(N_NODES), 256, 0, stream>>>(den2, N_NODES, 0.f);
    k_fill_f32<<<gridFor((long long)N_GRAPHS * OUT_DIM), 256, 0, stream>>>(psum, (long long)N_GRAPHS * OUT_DIM, 0.f);
    k_fill_f32<<<gridFor(N_GRAPHS), 256, 0, stream>>>(pcnt, N_GRAPHS, 0.f);
    k_fill_u32<<<gridFor((long long)N_NODES * HEADS), 256, 0, stream>>>(m1, (long long)N_NODES * HEADS, ENC_NEG_INF);
    k_fill_u32<<<gridFor(N_NODES), 256, 0, stream>>>(m2, N_NODES, ENC_NEG_INF);

    // ---- self-loop mean edge attrs + collapsed edge-weight matrices ----
    k_loop_attr_accum<<<gridFor(N_EDGES), 256, 0, stream>>>(dst, ea, la, cnt);
    k_loop_attr_div<<<gridFor(N_NODES), 256, 0, stream>>>(la, cnt);
    k_edge_mix<<<1, 32, 0, stream>>>(We1, ae1, We2, ae2, em);

    // ================= layer 1 (heads=4, dim=32) =================
    {
        int waves = (N_NODES / 16) * (128 / 16);
        k_wmma_gemm<128><<<(waves + 7) / 8, 256, 0, stream>>>(x, W1, xp1, N_NODES);
    }
    k_alphas<HEADS, HID><<<gridFor((long long)N_NODES * HEADS), 256, 0, stream>>>(xp1, as1w, ad1w, as1, ad1);
    k_edge_max<HEADS><<<gridFor((long long)E_TOT * HEADS), 256, 0, stream>>>(src, dst, ea, la, em, as1, ad1, m1);
    k_edge_sum<HEADS><<<gridFor((long long)E_TOT * HEADS), 256, 0, stream>>>(src, dst, ea, la, em, as1, ad1, m1, den1);
    k_edge_aggr<HEADS, HID><<<gridFor((long long)E_TOT * 128), 256, 0, stream>>>(src, dst, ea, la, em, as1, ad1, m1, den1, xp1, h1);
    k_bias_relu<<<gridFor((long long)N_NODES * 128), 256, 0, stream>>>(h1, b1, (long long)N_NODES * 128, 128);

    // ================= layer 2 (heads=1, dim=64) =================
    {
        int waves = (N_NODES / 16) * (64 / 16);
        k_wmma_gemm<64><<<(waves + 7) / 8, 256, 0, stream>>>(h1, W2, xp2, N_NODES);
    }
    k_alphas<1, OUT_DIM><<<gridFor(N_NODES), 256, 0, stream>>>(xp2, as2w, ad2w, as2, ad2);
    k_edge_max<1><<<gridFor((long long)E_TOT), 256, 0, stream>>>(src, dst, ea, la, em + 8, as2, ad2, m2);
    k_edge_sum<1><<<gridFor((long long)E_TOT), 256, 0, stream>>>(src, dst, ea, la, em + 8, as2, ad2, m2, den2);
    k_edge_aggr<1, OUT_DIM><<<gridFor((long long)E_TOT * 64), 256, 0, stream>>>(src, dst, ea, la, em + 8, as2, ad2, m2, den2, xp2, out2);

    // ================= global mean pool (+b2 fused) =================
    k_pool_accum<<<gridFor((long long)N_NODES * OUT_DIM), 256, 0, stream>>>(out2, pm, psum, pcnt);
    k_pool_final<<<gridFor((long long)N_GRAPHS * OUT_DIM), 256, 0, stream>>>(psum, pcnt, b2, (float*)d_out);
}